// AssignModule_72310069395899
// MI455X (gfx1250) — compile-verified
//
#include <hip/hip_runtime.h>
#include <hip/hip_bf16.h>
#include <stdint.h>

// ---------------------------------------------------------------------------
// Voxel seeding for MI455X (gfx1250, wave32).
// Pipeline: bbox (wave-uniform shuffle reduce + LDS atomics) -> 8x bisection
// (hash-set unique counts, async-to-LDS double-buffered streaming) -> final
// voxelization (centroid + argmin, global atomics) -> hybrid LDS/global
// bitonic sort of unique keys -> emit.
// ---------------------------------------------------------------------------

#define GCOMP     64
#define MASK40    ((1ULL << 40) - 1ULL)
#define EMPTY64   0xFFFFFFFFFFFFFFFFULL
#define TILE_PTS  1024          // points staged per tile (12 KB)
#define CTPB      256           // threads per block in count kernel
#define BSORT     4096          // elements per LDS bitonic window (32 KB)

// ---- order-preserving float <-> uint encoding (for atomic min/max) --------
__device__ __forceinline__ unsigned fenc(float f) {
  unsigned u = __float_as_uint(f);
  return (u & 0x80000000u) ? ~u : (u | 0x80000000u);
}
__device__ __forceinline__ float fdec(unsigned e) {
  unsigned u = (e & 0x80000000u) ? (e & 0x7FFFFFFFu) : ~e;
  return __uint_as_float(u);
}

// ---- 64-bit mix for hash-table slot ---------------------------------------
__device__ __forceinline__ unsigned hmix(unsigned long long k) {
  k ^= k >> 33; k *= 0xff51afd7ed558ccdULL;
  k ^= k >> 33; k *= 0xc4ceb9fe1a85ec53ULL;
  k ^= k >> 33;
  return (unsigned)k;
}

// ---- voxel key exactly as reference: (comp<<40) | (spatial_hash & MASK40) -
__device__ __forceinline__ unsigned long long voxel_key(
    float x, float y, float z, int c, const float* minsf, float v) {
  float mx = minsf[c * 3 + 0], my = minsf[c * 3 + 1], mz = minsf[c * 3 + 2];
  long long gx = (long long)floorf((x - mx) / v);
  long long gy = (long long)floorf((y - my) / v);
  long long gz = (long long)floorf((z - mz) / v);
  long long h = gx * 73856093LL ^ gy * 19349663LL ^ gz * 83492791LL;
  return ((unsigned long long)c << 40) | ((unsigned long long)h & MASK40);
}

// ---- open-addressing hash set / map ---------------------------------------
__device__ __forceinline__ int table_insert(unsigned long long* tk, unsigned mask,
                                            unsigned long long key, unsigned* slot) {
  unsigned idx = hmix(key) & mask;
  for (unsigned p = 0; p <= mask; ++p) {
    unsigned long long prev = atomicCAS(&tk[idx], EMPTY64, key);
    if (prev == EMPTY64) { *slot = idx; return 1; }
    if (prev == key)     { *slot = idx; return 0; }
    idx = (idx + 1) & mask;
  }
  *slot = idx & mask;
  return 0;
}
__device__ __forceinline__ unsigned table_find(const unsigned long long* tk,
                                               unsigned mask, unsigned long long key) {
  unsigned idx = hmix(key) & mask;
  for (unsigned p = 0; p <= mask; ++p) {
    unsigned long long v = tk[idx];
    if (v == key || v == EMPTY64) return idx;
    idx = (idx + 1) & mask;
  }
  return idx & mask;
}

// ---- CDNA5 async copy: global -> LDS, 16 B per lane, ASYNCcnt-tracked ------
__device__ __forceinline__ void async_b128(unsigned lds_off, const void* gaddr) {
  asm volatile("global_load_async_to_lds_b128 %0, %1, off"
               :: "v"(lds_off), "v"((unsigned long long)(uintptr_t)gaddr)
               : "memory");
}
#define ASYNC_WAIT_3() asm volatile("s_wait_asynccnt 3" ::: "memory")
#define ASYNC_WAIT_0() asm volatile("s_wait_asynccnt 0" ::: "memory")

// ===========================================================================
// Kernels
// ===========================================================================

__global__ __launch_bounds__(256) void k_init_small(unsigned* minsE, unsigned* maxsE,
                                                    unsigned* nper, unsigned* ucount) {
  int i = threadIdx.x;
  if (i < GCOMP * 3) { minsE[i] = 0xFFFFFFFFu; maxsE[i] = 0u; }
  if (i < GCOMP) nper[i] = 0u;
  if (i == 0) *ucount = 0u;
}

// Per-component bbox + counts. comp_lid is sorted, so a wave32 is almost
// always single-component: detect with ballot, reduce with shfl_xor (5 steps),
// one DS atomic set per wave instead of 32.
__global__ __launch_bounds__(256) void k_bbox(const float* __restrict__ xyz,
                                              const int* __restrict__ comp,
                                              unsigned* __restrict__ minsE,
                                              unsigned* __restrict__ maxsE,
                                              unsigned* __restrict__ nper,
                                              long long n) {
  __shared__ unsigned smin[GCOMP * 3], smax[GCOMP * 3], scnt[GCOMP];
  int tid = threadIdx.x;
  for (int i = tid; i < GCOMP * 3; i += blockDim.x) { smin[i] = 0xFFFFFFFFu; smax[i] = 0u; }
  for (int i = tid; i < GCOMP; i += blockDim.x) scnt[i] = 0u;
  __syncthreads();
  long long stride = (long long)gridDim.x * blockDim.x;
  for (long long i = (long long)blockIdx.x * blockDim.x + tid; i < n; i += stride) {
    float x = xyz[3 * i + 0], y = xyz[3 * i + 1], z = xyz[3 * i + 2];
    int c = comp[i];
    unsigned long long act = __ballot(1);
    bool full = (act == 0xFFFFFFFFull);
    bool uni  = full && (__ballot(c == __shfl(c, 0)) == 0xFFFFFFFFull);
    if (uni) {
      unsigned ex = fenc(x), ey = fenc(y), ez = fenc(z);
      unsigned nx = ex, ny = ey, nz = ez;   // mins
      unsigned Xx = ex, Xy = ey, Xz = ez;   // maxs
#pragma unroll
      for (int o = 16; o > 0; o >>= 1) {
        nx = min(nx, (unsigned)__shfl_xor((int)nx, o));
        ny = min(ny, (unsigned)__shfl_xor((int)ny, o));
        nz = min(nz, (unsigned)__shfl_xor((int)nz, o));
        Xx = max(Xx, (unsigned)__shfl_xor((int)Xx, o));
        Xy = max(Xy, (unsigned)__shfl_xor((int)Xy, o));
        Xz = max(Xz, (unsigned)__shfl_xor((int)Xz, o));
      }
      if ((tid & 31) == 0) {
        atomicMin(&smin[c * 3 + 0], nx); atomicMax(&smax[c * 3 + 0], Xx);
        atomicMin(&smin[c * 3 + 1], ny); atomicMax(&smax[c * 3 + 1], Xy);
        atomicMin(&smin[c * 3 + 2], nz); atomicMax(&smax[c * 3 + 2], Xz);
        atomicAdd(&scnt[c], 32u);
      }
    } else {
      atomicMin(&smin[c * 3 + 0], fenc(x)); atomicMax(&smax[c * 3 + 0], fenc(x));
      atomicMin(&smin[c * 3 + 1], fenc(y)); atomicMax(&smax[c * 3 + 1], fenc(y));
      atomicMin(&smin[c * 3 + 2], fenc(z)); atomicMax(&smax[c * 3 + 2], fenc(z));
      atomicAdd(&scnt[c], 1u);
    }
  }
  __syncthreads();
  for (int i = tid; i < GCOMP * 3; i += blockDim.x) {
    atomicMin(&minsE[i], smin[i]);
    atomicMax(&maxsE[i], smax[i]);
  }
  for (int i = tid; i < GCOMP; i += blockDim.x) atomicAdd(&nper[i], scnt[i]);
}

__global__ void k_setup(const unsigned* __restrict__ minsE, const unsigned* __restrict__ maxsE,
                        const unsigned* __restrict__ nper, const int* __restrict__ salloc,
                        float* __restrict__ minsf, long long* __restrict__ sc,
                        float* __restrict__ vlo, float* __restrict__ vhi,
                        float* __restrict__ bestv, long long* __restrict__ bestdiff) {
  int g = threadIdx.x;
  if (g >= GCOMP) return;
  float sp[3];
  for (int d = 0; d < 3; ++d) {
    float mn = fdec(minsE[g * 3 + d]);
    float mx = fdec(maxsE[g * 3 + d]);
    minsf[g * 3 + d] = mn;
    float s = mx - mn;
    s = fmaxf(s, 1e-6f);     // clip(span, 1e-6)
    s = fmaxf(s, 0.05f);     // safe_span
    sp[d] = s;
  }
  float vol = sp[0] * sp[1] * sp[2];
  long long np = (long long)nper[g];
  long long s = (long long)salloc[g];
  if (s < 1) s = 1;
  long long cap = np > 1 ? np : 1;
  if (s > cap) s = cap;
  sc[g] = s;
  float v0 = cbrtf(vol / fmaxf((float)s, 1.0f));
  vlo[g] = fmaxf(v0 * 0.1f, 1e-4f);
  vhi[g] = fmaxf(v0 * 64.0f, 1e-4f);
  bestv[g] = v0;
  bestdiff[g] = 1LL << 30;
}

__global__ __launch_bounds__(256) void k_clear_iter(unsigned long long* __restrict__ tkeys,
                                                    unsigned* __restrict__ cntG, long long C) {
  long long i = (long long)blockIdx.x * blockDim.x + threadIdx.x;
  if (i < C) tkeys[i] = EMPTY64;
  if (i < GCOMP) cntG[i] = 0u;
}

// Unique-voxel counting pass: double-buffered async-to-LDS staging of xyz,
// overlapping the next tile's fetch with the current tile's table probes.
// Per-component params cached in LDS (v_mid precomputed).
__global__ __launch_bounds__(CTPB) void k_count(const float* __restrict__ xyz,
                                                const int* __restrict__ comp,
                                                const float* __restrict__ minsf,
                                                const float* __restrict__ vlo,
                                                const float* __restrict__ vhi,
                                                unsigned long long* __restrict__ tkeys,
                                                unsigned* __restrict__ cntG,
                                                unsigned mask, long long n) {
  __shared__ float sbuf[2][TILE_PTS * 3];   // 2 x 12 KB
  __shared__ float sminc[GCOMP * 3];
  __shared__ float svmid[GCOMP];
  const int tid = threadIdx.x;
  if (tid < GCOMP * 3) sminc[tid] = minsf[tid];
  if (tid < GCOMP) svmid[tid] = (vlo[tid] + vhi[tid]) * 0.5f;
  __syncthreads();

  const long long nt = n / TILE_PTS;
  long long t = blockIdx.x;
  if (t < nt) {
    {  // stage first tile into buffer 0
      const char* g = (const char*)xyz + t * (TILE_PTS * 12LL);
      unsigned lbase = (unsigned)(uintptr_t)(&sbuf[0][0]);
#pragma unroll
      for (int c = 0; c < 3; ++c) {
        unsigned off = (unsigned)((c * CTPB + tid) * 16);
        async_b128(lbase + off, g + off);
      }
    }
    int buf = 0;
    while (t < nt) {
      long long nxt = t + gridDim.x;
      if (nxt < nt) {
        const char* g = (const char*)xyz + nxt * (TILE_PTS * 12LL);
        unsigned lbase = (unsigned)(uintptr_t)(&sbuf[buf ^ 1][0]);
#pragma unroll
        for (int c = 0; c < 3; ++c) {
          unsigned off = (unsigned)((c * CTPB + tid) * 16);
          async_b128(lbase + off, g + off);
        }
        __builtin_prefetch(&comp[nxt * TILE_PTS + tid], 0, 0);  // global_prefetch_b8
        ASYNC_WAIT_3();   // current tile's 3 loads done; next tile in flight
      } else {
        ASYNC_WAIT_0();
      }
      __syncthreads();    // all waves' stages for current buffer complete
      const float* s = &sbuf[buf][0];
      long long pbase = t * TILE_PTS;
      for (int p = tid; p < TILE_PTS; p += CTPB) {
        long long i = pbase + p;
        float x = s[3 * p + 0], y = s[3 * p + 1], z = s[3 * p + 2];
        int c = comp[i];
        unsigned long long key = voxel_key(x, y, z, c, sminc, svmid[c]);
        unsigned slot;
        if (table_insert(tkeys, mask, key, &slot)) atomicAdd(&cntG[c], 1u);
      }
      __syncthreads();    // everyone done reading buf before it is restaged
      buf ^= 1;
      t = nxt;
    }
  }
  // tail (n not a multiple of TILE_PTS); plain path
  long long stride = (long long)gridDim.x * blockDim.x;
  for (long long i = nt * TILE_PTS + (long long)blockIdx.x * blockDim.x + tid; i < n; i += stride) {
    float x = xyz[3 * i], y = xyz[3 * i + 1], z = xyz[3 * i + 2];
    int c = comp[i];
    unsigned long long key = voxel_key(x, y, z, c, sminc, svmid[c]);
    unsigned slot;
    if (table_insert(tkeys, mask, key, &slot)) atomicAdd(&cntG[c], 1u);
  }
}

__global__ void k_update(float* __restrict__ vlo, float* __restrict__ vhi,
                         float* __restrict__ bestv, long long* __restrict__ bestdiff,
                         const unsigned* __restrict__ cntG, const long long* __restrict__ sc) {
  int g = threadIdx.x;
  if (g >= GCOMP) return;
  float lo = vlo[g], hi = vhi[g];
  float mid = (lo + hi) * 0.5f;
  long long cnt = (long long)cntG[g];
  long long s = sc[g];
  long long diff = cnt > s ? cnt - s : s - cnt;
  if (diff < bestdiff[g]) { bestdiff[g] = diff; bestv[g] = mid; }
  if (cnt > s) vlo[g] = mid; else vhi[g] = mid;
}

__global__ __launch_bounds__(256) void k_clear_final(unsigned long long* __restrict__ tkeys,
                                                     unsigned* __restrict__ tcnt,
                                                     float* __restrict__ sx, float* __restrict__ sy,
                                                     float* __restrict__ sz,
                                                     unsigned* __restrict__ dmin,
                                                     unsigned* __restrict__ targ, long long C) {
  long long i = (long long)blockIdx.x * blockDim.x + threadIdx.x;
  if (i >= C) return;
  tkeys[i] = EMPTY64;
  tcnt[i] = 0u;
  sx[i] = 0.f; sy[i] = 0.f; sz[i] = 0.f;
  dmin[i] = 0xFFFFFFFFu;
  targ[i] = 0xFFFFFFFFu;
}

__global__ __launch_bounds__(256) void k_fill64(unsigned long long* __restrict__ a, long long n) {
  long long i = (long long)blockIdx.x * blockDim.x + threadIdx.x;
  if (i < n) a[i] = EMPTY64;
}

__global__ __launch_bounds__(256) void k_assign(const float* __restrict__ xyz,
                                                const int* __restrict__ comp,
                                                const float* __restrict__ minsf,
                                                const float* __restrict__ bestv,
                                                unsigned long long* __restrict__ tkeys,
                                                unsigned* __restrict__ tcnt,
                                                float* __restrict__ sx, float* __restrict__ sy,
                                                float* __restrict__ sz,
                                                unsigned* __restrict__ ptslot,
                                                unsigned mask, long long n) {
  __shared__ float sminc[GCOMP * 3];
  __shared__ float sbv[GCOMP];
  int tid = threadIdx.x;
  if (tid < GCOMP * 3) sminc[tid] = minsf[tid];
  if (tid < GCOMP) sbv[tid] = bestv[tid];
  __syncthreads();
  long long stride = (long long)gridDim.x * blockDim.x;
  for (long long i = (long long)blockIdx.x * blockDim.x + tid; i < n; i += stride) {
    float x = xyz[3 * i], y = xyz[3 * i + 1], z = xyz[3 * i + 2];
    int c = comp[i];
    unsigned long long key = voxel_key(x, y, z, c, sminc, sbv[c]);
    unsigned slot;
    table_insert(tkeys, mask, key, &slot);
    ptslot[i] = slot;
    atomicAdd(&tcnt[slot], 1u);
    atomicAdd(&sx[slot], x);
    atomicAdd(&sy[slot], y);
    atomicAdd(&sz[slot], z);
  }
}

__global__ __launch_bounds__(256) void k_dist(const float* __restrict__ xyz,
                                              const unsigned* __restrict__ ptslot,
                                              const unsigned* __restrict__ tcnt,
                                              const float* __restrict__ sx,
                                              const float* __restrict__ sy,
                                              const float* __restrict__ sz,
                                              unsigned* __restrict__ dmin,
                                              unsigned* __restrict__ pdbits, long long n) {
  long long stride = (long long)gridDim.x * blockDim.x;
  for (long long i = (long long)blockIdx.x * blockDim.x + threadIdx.x; i < n; i += stride) {
    unsigned slot = ptslot[i];
    unsigned c = tcnt[slot];
    float inv = 1.0f / (float)(c > 0u ? c : 1u);
    float cx = sx[slot] * inv, cy = sy[slot] * inv, cz = sz[slot] * inv;
    float dx = xyz[3 * i] - cx, dy = xyz[3 * i + 1] - cy, dz = xyz[3 * i + 2] - cz;
    float d = dx * dx + dy * dy + dz * dz;           // >= 0: uint bits order == float order
    unsigned bits = __float_as_uint(d);
    pdbits[i] = bits;
    atomicMin(&dmin[slot], bits);
  }
}

__global__ __launch_bounds__(256) void k_argmin(const unsigned* __restrict__ ptslot,
                                                const unsigned* __restrict__ pdbits,
                                                const unsigned* __restrict__ dmin,
                                                unsigned* __restrict__ targ, long long n) {
  long long stride = (long long)gridDim.x * blockDim.x;
  for (long long i = (long long)blockIdx.x * blockDim.x + threadIdx.x; i < n; i += stride) {
    unsigned slot = ptslot[i];
    if (pdbits[i] <= dmin[slot]) atomicMin(&targ[slot], (unsigned)i);
  }
}

__global__ __launch_bounds__(256) void k_collect(const unsigned long long* __restrict__ tkeys,
                                                 long long C,
                                                 unsigned long long* __restrict__ skeys,
                                                 unsigned* __restrict__ ucount) {
  long long i = (long long)blockIdx.x * blockDim.x + threadIdx.x;
  if (i >= C) return;
  unsigned long long k = tkeys[i];
  if (k != EMPTY64) {
    unsigned j = atomicAdd(ucount, 1u);
    skeys[j] = k;
  }
}

// Global bitonic pass (only for strides j >= BSORT).
__global__ __launch_bounds__(256) void k_bitonic(unsigned long long* __restrict__ a,
                                                 unsigned j, unsigned k, unsigned nsz) {
  unsigned i = blockIdx.x * blockDim.x + threadIdx.x;
  if (i >= nsz) return;
  unsigned ixj = i ^ j;
  if (ixj > i) {
    unsigned long long x = a[i], y = a[ixj];
    bool up = ((i & k) == 0u);
    if ((x > y) == up) { a[i] = y; a[ixj] = x; }
  }
}

// LDS-resident bitonic passes: performs stages k2 = kstart..kend with all
// strides j <= BSORT/2 done in LDS (one global read + one global write per
// BSORT-element window instead of one per pass).
__global__ __launch_bounds__(256) void k_bsort_local(unsigned long long* __restrict__ a,
                                                     unsigned kstart, unsigned kend) {
  __shared__ unsigned long long sk[BSORT];   // 32 KB
  const unsigned tid = threadIdx.x;
  const unsigned base = blockIdx.x * BSORT;
  for (unsigned e = tid; e < BSORT; e += 256) sk[e] = a[base + e];
  for (unsigned k2 = kstart; k2 != 0 && k2 <= kend; k2 <<= 1) {
    unsigned j0 = (k2 >> 1) < (BSORT / 2) ? (k2 >> 1) : (BSORT / 2);
    for (unsigned j = j0; j >= 1; j >>= 1) {
      __syncthreads();
      for (unsigned t = tid; t < BSORT / 2; t += 256) {
        unsigned i = ((t & ~(j - 1)) << 1) | (t & (j - 1));
        unsigned gi = base + i;
        bool up = ((gi & k2) == 0u);
        unsigned long long x = sk[i], y = sk[i + j];
        if ((x > y) == up) { sk[i] = y; sk[i + j] = x; }
      }
    }
    if (k2 == kend) break;   // avoid 32-bit overflow when kend is large
  }
  __syncthreads();
  for (unsigned e = tid; e < BSORT; e += 256) a[base + e] = sk[e];
}

__global__ __launch_bounds__(256) void k_emit(const unsigned long long* __restrict__ skeys,
                                              const unsigned long long* __restrict__ tkeys,
                                              const unsigned* __restrict__ targ,
                                              const float* __restrict__ xyz,
                                              float* __restrict__ out,
                                              unsigned mask, long long n) {
  long long j = (long long)blockIdx.x * blockDim.x + threadIdx.x;
  if (j >= n) return;
  float* oxyz = out;
  float* ogid = out + 3 * n;
  float* oval = out + 4 * n;
  unsigned long long key = skeys[j];
  if (key == EMPTY64) {
    oxyz[3 * j + 0] = 0.f; oxyz[3 * j + 1] = 0.f; oxyz[3 * j + 2] = 0.f;
    ogid[j] = -1.f;
    oval[j] = 0.f;
  } else {
    unsigned slot = table_find(tkeys, mask, key);
    unsigned a = targ[slot];
    if (a > (unsigned)(n - 1)) a = (unsigned)(n - 1);
    oxyz[3 * j + 0] = xyz[3 * a + 0];
    oxyz[3 * j + 1] = xyz[3 * a + 1];
    oxyz[3 * j + 2] = xyz[3 * a + 2];
    ogid[j] = (float)a;
    oval[j] = 1.f;
  }
}

// ===========================================================================
// Host launcher
// ===========================================================================
extern "C" void kernel_launch(void* const* d_in, const int* in_sizes, int n_in,
                              void* d_out, int out_size, void* d_ws, size_t ws_size,
                              hipStream_t stream) {
  (void)n_in; (void)out_size;
  const float* xyz   = (const float*)d_in[0];
  const int* comp    = (const int*)d_in[1];
  const int* salloc  = (const int*)d_in[2];
  const long long n  = (long long)in_sizes[1];
  float* out = (float*)d_out;

  long long NS = 1;
  while (NS < n) NS <<= 1;

  // --- carve workspace ---
  char* base = (char*)d_ws;
  size_t off = 0;
  auto carve = [&](size_t bytes, size_t align) -> void* {
    off = (off + align - 1) & ~(align - 1);
    void* p = base + off;
    off += bytes;
    return p;
  };
  unsigned* minsE      = (unsigned*)carve(GCOMP * 3 * 4, 256);
  unsigned* maxsE      = (unsigned*)carve(GCOMP * 3 * 4, 4);
  unsigned* nper       = (unsigned*)carve(GCOMP * 4, 4);
  unsigned* cntG       = (unsigned*)carve(GCOMP * 4, 4);
  unsigned* ucount     = (unsigned*)carve(4, 4);
  float* minsf         = (float*)carve(GCOMP * 3 * 4, 4);
  long long* scA       = (long long*)carve(GCOMP * 8, 8);
  long long* bestdiff  = (long long*)carve(GCOMP * 8, 8);
  float* vlo           = (float*)carve(GCOMP * 4, 4);
  float* vhi           = (float*)carve(GCOMP * 4, 4);
  float* bestv         = (float*)carve(GCOMP * 4, 4);
  unsigned* ptslot     = (unsigned*)carve((size_t)n * 4, 256);
  unsigned* pdbits     = (unsigned*)carve((size_t)n * 4, 256);
  unsigned long long* skeys = (unsigned long long*)carve((size_t)NS * 8, 256);

  // Hash table: largest power of two that fits; capped at 4M slots so the key
  // array (32 MB) + points (24 MB) stay resident in the 192 MB L2.
  size_t rem  = (ws_size > off + 512) ? (ws_size - off - 512) : 0;
  size_t maxC = rem / 32;    // 8(key)+4(cnt)+12(sum)+4(dmin)+4(argmin)
  size_t C = 1024;
  while ((C << 1) <= maxC && C < (1ULL << 22)) C <<= 1;
  unsigned mask = (unsigned)(C - 1);
  unsigned long long* tkeys = (unsigned long long*)carve(C * 8, 256);
  unsigned* tcnt = (unsigned*)carve(C * 4, 256);
  float* sx      = (float*)carve(C * 4, 256);
  float* sy      = (float*)carve(C * 4, 256);
  float* sz      = (float*)carve(C * 4, 256);
  unsigned* dmin = (unsigned*)carve(C * 4, 256);
  unsigned* targ = (unsigned*)carve(C * 4, 256);

  int gN  = (int)((n + 255) / 256);
  int gC  = (int)((C + 255) / 256);
  int gNS = (int)((NS + 255) / 256);

  k_init_small<<<1, 256, 0, stream>>>(minsE, maxsE, nper, ucount);
  k_bbox<<<1024, 256, 0, stream>>>(xyz, comp, minsE, maxsE, nper, n);
  k_setup<<<1, 64, 0, stream>>>(minsE, maxsE, nper, salloc, minsf, scA, vlo, vhi, bestv, bestdiff);

  for (int it = 0; it < 8; ++it) {
    k_clear_iter<<<gC, 256, 0, stream>>>(tkeys, cntG, (long long)C);
    k_count<<<256, CTPB, 0, stream>>>(xyz, comp, minsf, vlo, vhi, tkeys, cntG, mask, n);
    k_update<<<1, 64, 0, stream>>>(vlo, vhi, bestv, bestdiff, cntG, scA);
  }

  k_clear_final<<<gC, 256, 0, stream>>>(tkeys, tcnt, sx, sy, sz, dmin, targ, (long long)C);
  k_fill64<<<gNS, 256, 0, stream>>>(skeys, NS);
  k_assign<<<2048, 256, 0, stream>>>(xyz, comp, minsf, bestv, tkeys, tcnt, sx, sy, sz, ptslot, mask, n);
  k_dist<<<2048, 256, 0, stream>>>(xyz, ptslot, tcnt, sx, sy, sz, dmin, pdbits, n);
  k_argmin<<<2048, 256, 0, stream>>>(ptslot, pdbits, dmin, targ, n);
  k_collect<<<gC, 256, 0, stream>>>(tkeys, (long long)C, skeys, ucount);

  // ---- hybrid bitonic sort of unique keys (ascending, EMPTY64 sorts last) --
  if (NS >= (long long)BSORT) {
    int gB = (int)(NS / BSORT);
    // all stages k = 2..BSORT fully inside LDS windows
    k_bsort_local<<<gB, 256, 0, stream>>>(skeys, 2u, (unsigned)BSORT);
    for (unsigned long long k = (unsigned long long)BSORT * 2; k <= (unsigned long long)NS; k <<= 1) {
      for (unsigned long long j = k >> 1; j >= (unsigned long long)BSORT; j >>= 1)
        k_bitonic<<<gNS, 256, 0, stream>>>(skeys, (unsigned)j, (unsigned)k, (unsigned)NS);
      // strides BSORT/2 .. 1 of this stage fused into one LDS kernel
      k_bsort_local<<<gB, 256, 0, stream>>>(skeys, (unsigned)k, (unsigned)k);
    }
  } else {
    for (unsigned long long k = 2; k <= (unsigned long long)NS; k <<= 1)
      for (unsigned long long j = k >> 1; j > 0; j >>= 1)
        k_bitonic<<<gNS, 256, 0, stream>>>(skeys, (unsigned)j, (unsigned)k, (unsigned)NS);
  }

  k_emit<<<gN, 256, 0, stream>>>(skeys, tkeys, targ, xyz, out, mask, n);
}